// EmbedBigraph_86346022518981
// MI455X (gfx1250) — compile-verified
//
#include <hip/hip_runtime.h>

// EmbedBigraph: per-target-node masked gather + LayerNorm + per-node GEMM.
// Memory-bound by streaming W (1.07 GB) once; x (16.8 MB) is L2-resident.
// Matrix core path: V_WMMA_F32_16X16X4_F32 (fp32, faithful to reference).

typedef __attribute__((ext_vector_type(2))) float v2f;
typedef __attribute__((ext_vector_type(8))) float v8f;

#define N_B    16
#define C_IN   128
#define C_OUT  128
#define M_SRC  2048
#define L_TGT  1024
#define KMAX   16
#define CKDIM  (C_IN * KMAX)   // 2048 = GEMM K dimension
#define EPSV   1e-5f

__launch_bounds__(256, 2)
__global__ void embed_bigraph_kernel(const float* __restrict__ x,      // (N, C_IN, M)
                                     const float* __restrict__ W,      // (L, C_OUT, C_IN, K)
                                     const float* __restrict__ bias,   // (L, C_OUT)
                                     const int*   __restrict__ idx,    // (L, K)
                                     const unsigned char* __restrict__ mask, // (L, K) bool (1B)
                                     float* __restrict__ out)          // (N, C_OUT, L)
{
    // Zb holds zn_l (2048 x 16) in WMMA-B interleaved order:
    // element (ck, n) at float offset (ck>>1)*32 + n*2 + (ck&1).
    // A wave lane then fetches its whole B fragment with one 8B ds load,
    // hitting all 64 LDS banks exactly once across the wave.
    __shared__ float Zb[CKDIM * N_B];        // 128 KB
    __shared__ float redS[256];
    __shared__ float redQ[256];
    __shared__ float s_mean[N_B];
    __shared__ float s_rsig[N_B];
    __shared__ int   s_idx[KMAX];
    __shared__ float s_msk[KMAX];

    const int l   = blockIdx.x;
    const int tid = threadIdx.x;

    if (tid < KMAX) {
        s_idx[tid] = idx[l * KMAX + tid];
        s_msk[tid] = mask[l * KMAX + tid] ? 1.0f : 0.0f;
    }
    __syncthreads();

    // ---------- Phase 1: masked gather -> LDS + per-thread partial stats ----------
    const int n = tid & 15;    // batch index this thread serves
    const int g = tid >> 4;    // 16 groups; group g covers ck in [g*128, g*128+128)
    const float* xn = x + (size_t)n * (C_IN * M_SRC);

    float psum = 0.0f, psq = 0.0f;
    #pragma unroll 4
    for (int i = 0; i < 128; ++i) {
        const int ck = g * 128 + i;
        const int c  = ck >> 4;
        const int k  = ck & 15;
        const float mk = s_msk[k];
        // padded taps have idx==0 (valid address), mk==0 zeroes them
        const float v = mk * xn[c * M_SRC + s_idx[k]];
        psum += v;
        psq  += v * v;
        Zb[(ck >> 1) * 32 + n * 2 + (ck & 1)] = v;
    }
    redS[tid] = psum;
    redQ[tid] = psq;
    __syncthreads();

    // ---------- Phase 2: reduce stats per n, masked-count LayerNorm params ----------
    if (tid < N_B) {
        float ss = 0.0f, qq = 0.0f;
        #pragma unroll
        for (int gg = 0; gg < 16; ++gg) {
            ss += redS[gg * 16 + tid];
            qq += redQ[gg * 16 + tid];
        }
        float nvalid = 0.0f;
        #pragma unroll
        for (int k = 0; k < KMAX; ++k) nvalid += s_msk[k];
        const float cnt  = (float)C_IN * nvalid;
        const float mean = ss / cnt;
        const float var  = qq / cnt - mean * mean;
        s_mean[tid] = mean;
        s_rsig[tid] = rsqrtf(var + EPSV);
    }
    __syncthreads();

    // ---------- Phase 3: normalize in place (mask keeps padded taps at 0) ----------
    {
        const float mean = s_mean[n];
        const float rsig = s_rsig[n];
        #pragma unroll 4
        for (int i = 0; i < 128; ++i) {
            const int ck = g * 128 + i;
            const int o  = (ck >> 1) * 32 + n * 2 + (ck & 1);
            Zb[o] = s_msk[ck & 15] * (Zb[o] - mean) * rsig;
        }
    }
    __syncthreads();

    // ---------- Phase 4: per-node GEMM, D(128x16) = W_l(128x2048) * Zb(2048x16) ----------
    // 8 waves; wave w owns C_OUT rows [16w, 16w+16). 512 chained f32 WMMAs, K=4 each.
    const int lane = tid & 31;
    const int wv   = tid >> 5;
    const int half = lane >> 4;       // 0: K pair {0,1}; 1: K pair {2,3}
    const int mrow = lane & 15;

    // A fragment source: lane streams its W row contiguously (coalesced over kk)
    const float* Ap = W + ((size_t)l * C_OUT + (wv * 16 + mrow)) * CKDIM + half * 2;
    // B fragment source in LDS
    const float* Bp = Zb + half * 32 + mrow * 2;

    v8f acc = {0.f, 0.f, 0.f, 0.f, 0.f, 0.f, 0.f, 0.f};
    #pragma unroll 8
    for (int kk = 0; kk < CKDIM; kk += 4) {
        const v2f a = *(const v2f*)(Ap + kk);                 // global_load_b64
        const v2f bb = *(const v2f*)(Bp + (kk >> 1) * 32);    // ds_load_b64, bank-conflict-free
        acc = __builtin_amdgcn_wmma_f32_16x16x4_f32(
            /*neg_a=*/false, a, /*neg_b=*/false, bb,
            /*c_mod=*/(short)0, acc, /*reuse_a=*/false, /*reuse_b=*/false);
    }

    // ---------- Phase 5: D layout scatter + bias ----------
    // VGPR r: lanes 0-15 -> (M=r, N=lane); lanes 16-31 -> (M=8+r, N=lane-16)
    #pragma unroll
    for (int r = 0; r < 8; ++r) {
        const int co = wv * 16 + half * 8 + r;                // C_OUT index
        const float bv = bias[l * C_OUT + co];
        out[((size_t)mrow * C_OUT + co) * L_TGT + l] = acc[r] + bv;
    }
}

extern "C" void kernel_launch(void* const* d_in, const int* in_sizes, int n_in,
                              void* d_out, int out_size, void* d_ws, size_t ws_size,
                              hipStream_t stream) {
    const float* x    = (const float*)d_in[0];
    const float* W    = (const float*)d_in[1];
    const float* b    = (const float*)d_in[2];
    const int*   idx  = (const int*)d_in[3];
    const unsigned char* mask = (const unsigned char*)d_in[4];  // jnp bool -> 1 byte
    float* out = (float*)d_out;

    embed_bigraph_kernel<<<dim3(L_TGT), dim3(256), 0, stream>>>(x, W, b, idx, mask, out);
}